// CrossModalSemanticGraph_40647570489402
// MI455X (gfx1250) — compile-verified
//
#include <hip/hip_runtime.h>
#include <math.h>

#define BN 4096
#define DD 512
#define NC 3
#define NT (BN + NC)          // 4099
#define DELTA_T 1.5f

typedef __attribute__((ext_vector_type(2))) float v2f;
typedef __attribute__((ext_vector_type(8))) float v8f;

// ---- workspace layout (float units) ----
#define CNT_OFF   0                        // counts[3][3]            (9)
#define CEN_OFF   16                       // centers[3][3*DD]        (4608)  sums -> means
#define AVG_OFF   (CEN_OFF + 3*3*DD)       // avg_centers[3*DD]       (1536)
#define SELF_OFF  (AVG_OFF + 3*DD)         // self[3][BN]             (12288)
#define LP_OFF    (SELF_OFF + 3*BN)        // lp[3][BN*3]             (36864)
#define CLS_OFF   (LP_OFF + 3*BN*3)        // cls[3][BN]  (int)       (12288)
#define E_OFF     (CLS_OFF + 3*BN)         // e[3][BN*3]              (36864)
#define PF_OFF    (E_OFF + 3*BN*3)         // pred_f[BN]  (int)       (4096)
#define WC_OFF    (PF_OFF + BN)            // w_center[BN]            (4096)

// ---------------- init: zero counts + center sums ----------------
__global__ void init_k(float* ws) {
    int i = blockIdx.x * 256 + threadIdx.x;
    if (i < CEN_OFF + 3*3*DD) ws[i] = 0.0f;
}

// ---------------- per-sample stats: pred, self-term, log p, counts ----------------
__global__ void stats_k(const float* __restrict__ p0, const float* __restrict__ p1,
                        const float* __restrict__ p2, float* ws) {
    int m = blockIdx.y;
    const float* P = (m == 0) ? p0 : ((m == 1) ? p1 : p2);
    int i = blockIdx.x * 256 + threadIdx.x;
    if (i >= BN) return;
    float a = P[i*3+0], b = P[i*3+1], c = P[i*3+2];
    float la = logf(a), lb = logf(b), lc = logf(c);
    float* LP = ws + LP_OFF + m * (BN*3);
    LP[i*3+0] = la; LP[i*3+1] = lb; LP[i*3+2] = lc;
    ws[SELF_OFF + m*BN + i] = a*la + b*lb + c*lc;
    int pred = 0; float best = a;
    if (b > best) { best = b; pred = 1; }
    if (c > best) { best = c; pred = 2; }
    ((int*)ws)[CLS_OFF + m*BN + i] = pred;
    atomicAdd(&ws[CNT_OFF + m*3 + pred], 1.0f);
}

// ---------------- segment-sum of features by class (atomic accumulate) ----------------
__global__ void center_accum_k(const float* __restrict__ z0, const float* __restrict__ z1,
                               const float* __restrict__ z2, float* ws) {
    int m = blockIdx.z;
    const float* Z = (m == 0) ? z0 : ((m == 1) ? z1 : z2);
    int d  = blockIdx.x * 256 + threadIdx.x;     // column, gridDim.x = 2
    int i0 = blockIdx.y * 256;                   // row chunk, gridDim.y = 16
    const int* cls = (const int*)ws + CLS_OFF + m*BN;
    float a0 = 0.f, a1 = 0.f, a2 = 0.f;
    for (int k = 0; k < 256; ++k) {
        int i = i0 + k;
        float v = Z[(size_t)i*DD + d];
        int c = cls[i];
        a0 += (c == 0) ? v : 0.f;
        a1 += (c == 1) ? v : 0.f;
        a2 += (c == 2) ? v : 0.f;
    }
    float* cen = ws + CEN_OFF + m*3*DD;
    atomicAdd(&cen[0*DD + d], a0);
    atomicAdd(&cen[1*DD + d], a1);
    atomicAdd(&cen[2*DD + d], a2);
}

// ---------------- finalize: sums -> means, avg_centers, node_feature center rows ----------------
__global__ void center_fin_k(float* ws, float* out) {
    int idx = blockIdx.x * 256 + threadIdx.x;    // over 3*DD
    if (idx >= 3*DD) return;
    int c = idx / DD, d = idx % DD;
    float s = 0.f;
    for (int m = 0; m < 3; ++m) {
        float cnt = ws[CNT_OFF + m*3 + c];
        float v = ws[CEN_OFF + m*3*DD + idx] / fmaxf(cnt, 1.0f);
        ws[CEN_OFF + m*3*DD + idx] = v;          // now a mean
        s += v;
    }
    float avg = s * (1.0f/3.0f);
    ws[AVG_OFF + idx] = avg;
    // node_features rows BN..BN+2
    out[(size_t)NT*NT + (size_t)(BN + c)*DD + d] = avg;
}

// ---------------- e[m][i][c] = exp(-0.5 ||z_i - center_c||^2), wave per sample ----------------
__global__ void dist_k(const float* __restrict__ z0, const float* __restrict__ z1,
                       const float* __restrict__ z2, float* ws) {
    int m = blockIdx.y;
    const float* Z = (m == 0) ? z0 : ((m == 1) ? z1 : z2);
    int wave = threadIdx.x >> 5, lane = threadIdx.x & 31;
    int i = blockIdx.x * 8 + wave;               // gridDim.x = BN/8
    const float* cen = ws + CEN_OFF + m*3*DD;
    float s0 = 0.f, s1 = 0.f, s2 = 0.f;
    for (int d = lane; d < DD; d += 32) {
        float zv = Z[(size_t)i*DD + d];
        float d0 = zv - cen[d];
        float d1 = zv - cen[DD + d];
        float d2 = zv - cen[2*DD + d];
        s0 += d0*d0; s1 += d1*d1; s2 += d2*d2;
    }
    for (int off = 16; off; off >>= 1) {
        s0 += __shfl_xor(s0, off, 32);
        s1 += __shfl_xor(s1, off, 32);
        s2 += __shfl_xor(s2, off, 32);
    }
    if (lane == 0) {
        float* E = ws + E_OFF + m*BN*3;
        E[i*3+0] = expf(-0.5f*s0);
        E[i*3+1] = expf(-0.5f*s1);
        E[i*3+2] = expf(-0.5f*s2);
    }
}

// ---------------- fused pseudo-label argmax + w_center, wave per sample ----------------
__global__ void fused_k(const float* __restrict__ fused, const float* __restrict__ ccp,
                        float* ws) {
    int wave = threadIdx.x >> 5, lane = threadIdx.x & 31;
    int i = blockIdx.x * 8 + wave;               // gridDim.x = BN/8
    float s0 = 0.f, s1 = 0.f, s2 = 0.f;
    for (int d = lane; d < DD; d += 32) {
        float f = fused[(size_t)i*DD + d];
        s0 += f * ccp[d];
        s1 += f * ccp[DD + d];
        s2 += f * ccp[2*DD + d];
    }
    for (int off = 16; off; off >>= 1) {
        s0 += __shfl_xor(s0, off, 32);
        s1 += __shfl_xor(s1, off, 32);
        s2 += __shfl_xor(s2, off, 32);
    }
    // softmax(logits/T) argmax == logits argmax
    int pf = 0; float best = s0;
    if (s1 > best) { best = s1; pf = 1; }
    if (s2 > best) { best = s2; pf = 2; }
    const float* avgc = ws + AVG_OFF + pf*DD;
    float t = 0.f;
    for (int d = lane; d < DD; d += 32) {
        float df = fused[(size_t)i*DD + d] - avgc[d];
        t += df*df;
    }
    for (int off = 16; off; off >>= 1) t += __shfl_xor(t, off, 32);
    if (lane == 0) {
        ((int*)ws)[PF_OFF + i] = pf;
        ws[WC_OFF + i] = expf(-0.5f*t);
    }
}

// ---------------- main B x B adjacency tiles via WMMA f32 16x16x4 ----------------
__global__ void __launch_bounds__(256) tile_k(const float* __restrict__ p0,
                                              const float* __restrict__ p1,
                                              const float* __restrict__ p2,
                                              const float* __restrict__ ws,
                                              float* __restrict__ out) {
    const int wave = threadIdx.x >> 5, lane = threadIdx.x & 31;
    const int tile = blockIdx.x * 8 + wave;      // 65536 tiles, gridDim.x = 8192
    const int ti = tile >> 8, tj = tile & 255;   // 256 tiles per dimension
    const int I0 = ti * 16, J0 = tj * 16;
    const int half = lane >> 4, r = lane & 15;

    float wsum[8];
#pragma unroll
    for (int v = 0; v < 8; ++v) wsum[v] = 0.0f;

    const v8f zc = {0.f,0.f,0.f,0.f,0.f,0.f,0.f,0.f};

#pragma unroll
    for (int m = 0; m < 3; ++m) {
        const float* Pm  = (m == 0) ? p0 : ((m == 1) ? p1 : p2);
        const float* LPm = ws + LP_OFF + m*BN*3;
        const float* Sm  = ws + SELF_OFF + m*BN;
        const int*   CLm = (const int*)ws + CLS_OFF + m*BN;
        const float* Em  = ws + E_OFF + m*BN*3;

        const int rowA = I0 + r;                 // A operand: rows of the tile
        const int colB = J0 + r;                 // B operand: cols of the tile

        // Unconditionally load the full 3-vector of each row (contiguous ->
        // global_load_b96), select K components in registers: no exec-mask
        // divergence before the WMMAs (EXEC stays all-ones).
        float pr0 = Pm[rowA*3+0], pr1 = Pm[rowA*3+1], pr2 = Pm[rowA*3+2];
        float lr0 = LPm[rowA*3+0], lr1 = LPm[rowA*3+1], lr2 = LPm[rowA*3+2];
        float pc0 = Pm[colB*3+0], pc1 = Pm[colB*3+1], pc2 = Pm[colB*3+2];
        float lc0 = LPm[colB*3+0], lc1 = LPm[colB*3+1], lc2 = LPm[colB*3+2];

        // A 16x4 f32 layout: lanes 0-15 hold K=0,1 ; lanes 16-31 hold K=2,3
        // (K=3 is the zero pad since C=3).
        v2f aP, aL, bP, bL;
        aP.x = half ? pr2 : pr0;  aP.y = half ? 0.0f : pr1;
        aL.x = half ? lr2 : lr0;  aL.y = half ? 0.0f : lr1;
        bP.x = half ? pc2 : pc0;  bP.y = half ? 0.0f : pc1;
        bL.x = half ? lc2 : lc0;  bL.y = half ? 0.0f : lc1;

        // crossA[i][j] = p_i . log p_j   ;   crossB[i][j] = p_j . log p_i
        v8f crossA = __builtin_amdgcn_wmma_f32_16x16x4_f32(
            false, aP, false, bL, (short)0, zc, false, false);
        v8f crossB = __builtin_amdgcn_wmma_f32_16x16x4_f32(
            false, aL, false, bP, (short)0, zc, false, false);

        const float sj  = Sm[J0 + r];
        const int   cj  = CLm[J0 + r];
        const float ej0 = Em[(J0 + r)*3 + 0];
        const float ej1 = Em[(J0 + r)*3 + 1];
        const float ej2 = Em[(J0 + r)*3 + 2];

#pragma unroll
        for (int v = 0; v < 8; ++v) {
            const int gi = I0 + v + 8*half;      // C/D layout: M = v + 8*(lane>=16)
            const float si = Sm[gi];
            const int   ci = CLm[gi];
            const float skl = 0.5f * (si + sj - crossA[v] - crossB[v]);
            const float ei = Em[gi*3 + cj];           // dm[i,j] = e[i, cls_j]
            const float ej = (ci == 0) ? ej0 : ((ci == 1) ? ej1 : ej2); // dm[j,i]
            const float w = (skl < DELTA_T)
                          ? (1.0f - skl * (1.0f/DELTA_T)) * ei * ej : 0.0f;
            wsum[v] += w;
        }
    }

#pragma unroll
    for (int v = 0; v < 8; ++v) {
        const int gi = I0 + v + 8*half;
        const int gj = J0 + r;
        const float val = (gi == gj) ? 0.0f : wsum[v] * (1.0f/3.0f);
        // streaming (non-temporal) store: adjacency is write-once, never re-read
        __builtin_nontemporal_store(val, &out[(size_t)gi * NT + gj]);
    }
}

// ---------------- borders: top-right / bottom-left strips + eye(3) ----------------
__global__ void border_k(const float* __restrict__ ws, float* __restrict__ out) {
    int i = blockIdx.x * 256 + threadIdx.x;
    if (i < BN) {
        float wc = ws[WC_OFF + i];
        int pf = ((const int*)ws)[PF_OFF + i];
#pragma unroll
        for (int c = 0; c < 3; ++c) {
            float v = (c == pf) ? wc : 0.0f;
            out[(size_t)i * NT + (BN + c)] = v;
            out[(size_t)(BN + c) * NT + i] = v;
        }
    }
    if (i < 9) {
        int rr = i / 3, cc = i % 3;
        out[(size_t)(BN + rr) * NT + (BN + cc)] = (rr == cc) ? 1.0f : 0.0f;
    }
}

extern "C" void kernel_launch(void* const* d_in, const int* in_sizes, int n_in,
                              void* d_out, int out_size, void* d_ws, size_t ws_size,
                              hipStream_t stream) {
    const float* zt  = (const float*)d_in[0];
    const float* za  = (const float*)d_in[1];
    const float* zv  = (const float*)d_in[2];
    const float* pt  = (const float*)d_in[3];
    const float* pa  = (const float*)d_in[4];
    const float* pv  = (const float*)d_in[5];
    const float* fus = (const float*)d_in[6];
    const float* ccp = (const float*)d_in[7];
    float* out = (float*)d_out;
    float* ws  = (float*)d_ws;

    init_k<<<(CEN_OFF + 3*3*DD + 255)/256, 256, 0, stream>>>(ws);
    stats_k<<<dim3(BN/256, 3), 256, 0, stream>>>(pt, pa, pv, ws);
    center_accum_k<<<dim3(DD/256, 16, 3), 256, 0, stream>>>(zt, za, zv, ws);
    center_fin_k<<<(3*DD + 255)/256, 256, 0, stream>>>(ws, out);
    dist_k<<<dim3(BN/8, 3), 256, 0, stream>>>(zt, za, zv, ws);
    fused_k<<<BN/8, 256, 0, stream>>>(fus, ccp, ws);
    tile_k<<<(256*256)/8, 256, 0, stream>>>(pt, pa, pv, ws, out);
    border_k<<<(BN + 255)/256, 256, 0, stream>>>(ws, out);
    // node_features rows 0..BN-1 = fused_representations
    hipMemcpyAsync(out + (size_t)NT*NT, fus, (size_t)BN*DD*sizeof(float),
                   hipMemcpyDeviceToDevice, stream);
}